// NestedGIN_37830071943189
// MI455X (gfx1250) — compile-verified
//
#include <hip/hip_runtime.h>

typedef __attribute__((ext_vector_type(2))) float v2f;
typedef __attribute__((ext_vector_type(8))) float v8f;

#define DFEAT 64

// ---------------------------------------------------------------- utilities
__global__ __launch_bounds__(256) void copy_f4(const float4* __restrict__ s,
                                               float4* __restrict__ d, int n) {
  int i = blockIdx.x * 256 + threadIdx.x;
  if (i < n) d[i] = s[i];
}

__global__ __launch_bounds__(256) void zero_f(float* __restrict__ p, int n) {
  int i = blockIdx.x * 256 + threadIdx.x;
  if (i < n) p[i] = 0.0f;
}

// ------------------------------------------------- edge scatter: agg += h[src]
// agg has been pre-initialized to h (fuses GIN's h + sum_{j->i} h_j, eps=0).
__global__ __launch_bounds__(256) void scatter_add(const float* __restrict__ h,
                                                   const int* __restrict__ esrc,
                                                   const int* __restrict__ edst,
                                                   float* __restrict__ agg,
                                                   int nedges) {
  long long gid = (long long)blockIdx.x * 256 + threadIdx.x;
  long long tot = (long long)nedges * 16;
  if (gid >= tot) return;
  int e = (int)(gid >> 4);
  int c = ((int)gid & 15) << 2;
  int s = esrc[e];
  int d = edst[e];
  const float* hp = h + (size_t)s * DFEAT + c;
  float* ap = agg + (size_t)d * DFEAT + c;
  float4 v = *(const float4*)hp;
  atomicAdd(ap + 0, v.x);
  atomicAdd(ap + 1, v.y);
  atomicAdd(ap + 2, v.z);
  atomicAdd(ap + 3, v.w);
}

// ---------------------------------------------- segment sum via f32 atomics
__global__ __launch_bounds__(256) void seg_sum(const float* __restrict__ in,
                                               const int* __restrict__ seg,
                                               float* __restrict__ out, int nrows) {
  int gid = blockIdx.x * 256 + threadIdx.x;
  if (gid >= nrows * DFEAT) return;
  int r = gid >> 6;
  int f = gid & 63;
  atomicAdd(out + (size_t)seg[r] * DFEAT + f, in[(size_t)r * DFEAT + f]);
}

// ---------------------------------------------------------------- WMMA GEMM
// Out[0:rows16,64] = act(Z @ W + bias) for rows16 a multiple of 16.
// One wave computes a 16-row M-tile across all 64 output columns using
// v_wmma_f32_16x16x4_f32 (exact fp32 accumulate): 64 WMMAs per tile.
// W is staged in LDS pre-swizzled into B-fragment order so each lane's
// B operand is a single aligned ds_load_b64 (no VGPR repack movs):
//   sWp[j*64 + n] = { W[2j][n], W[2j+1][n] },  j = 2*ks + lanehalf.
__global__ __launch_bounds__(256) void gemm64_wmma(const float* __restrict__ Z,
                                                   const float* __restrict__ W,
                                                   const float* __restrict__ bias,
                                                   float* __restrict__ Out,
                                                   int rows16, int relu) {
  __shared__ v2f sWp[32 * 64];
  __shared__ float sB[64];
  int t = threadIdx.x;
  for (int i = t; i < 2048; i += 256) {
    int j = i >> 6;
    int n = i & 63;
    v2f p;
    p.x = W[(2 * j) * DFEAT + n];
    p.y = W[(2 * j + 1) * DFEAT + n];
    sWp[i] = p;
  }
  if (t < 64) sB[t] = bias[t];
  __syncthreads();

  int lane = t & 31;
  int wave = t >> 5;
  int row0 = (blockIdx.x * 8 + wave) * 16;
  if (row0 >= rows16) return;  // wave-uniform: EXEC stays full for WMMA

  int lm = lane & 15;  // A: row-in-tile (M); B/C/D: column (N)
  int lh = lane >> 4;  // lane half selects K / M sub-block per ISA layout

  // A fragments: a.x = Z[M][4*ks + 2*lh], a.y = Z[M][4*ks + 2*lh + 1]
  const float* zrow = Z + (size_t)(row0 + lm) * DFEAT;
  v2f afrag[16];
#pragma unroll
  for (int ks = 0; ks < 16; ++ks) {
    float2 v = *(const float2*)(zrow + 4 * ks + 2 * lh);
    afrag[ks].x = v.x;
    afrag[ks].y = v.y;
  }

  const v2f* bbase = sWp + lh * 64 + lm;  // + ks*128 + nt*16 per fragment
#pragma unroll
  for (int nt = 0; nt < 4; ++nt) {
    v8f acc = {};
#pragma unroll
    for (int ks = 0; ks < 16; ++ks) {
      v2f b = bbase[ks * 128 + nt * 16];
      acc = __builtin_amdgcn_wmma_f32_16x16x4_f32(false, afrag[ks], false, b,
                                                  (short)0, acc, false, false);
    }
    float bv = sB[nt * 16 + lm];
#pragma unroll
    for (int v = 0; v < 8; ++v) {  // D: VGPR v holds M = v + 8*lh, N = lm
      float val = acc[v] + bv;
      if (relu) val = fmaxf(val, 0.0f);
      Out[(size_t)(row0 + v + 8 * lh) * DFEAT + nt * 16 + lm] = val;
    }
  }
}

// ----------------------------------------- scalar tail for rows % 16 != 0
__global__ __launch_bounds__(256) void gemm64_tail(const float* __restrict__ Z,
                                                   const float* __restrict__ W,
                                                   const float* __restrict__ bias,
                                                   float* __restrict__ Out,
                                                   int rstart, int rows, int relu) {
  int gid = blockIdx.x * 256 + threadIdx.x;
  int r = rstart + (gid >> 6);
  int n = gid & 63;
  if (r >= rows) return;
  const float* zrow = Z + (size_t)r * DFEAT;
  float acc = bias[n];
#pragma unroll
  for (int k = 0; k < DFEAT; ++k) acc = fmaf(zrow[k], W[k * DFEAT + n], acc);
  if (relu) acc = fmaxf(acc, 0.0f);
  Out[(size_t)r * DFEAT + n] = acc;
}

// ------------------------------------------------------------- output heads
__global__ __launch_bounds__(256) void head(const float* __restrict__ hid,
                                            const float* __restrict__ wr,
                                            const float* __restrict__ br,
                                            const float* __restrict__ wv,
                                            const float* __restrict__ bv,
                                            float* __restrict__ out, int ng) {
  int i = blockIdx.x * 256 + threadIdx.x;
  if (i >= ng) return;
  const float* hr = hid + (size_t)i * DFEAT;
  float sr = 0.f, sv = 0.f;
#pragma unroll
  for (int k = 0; k < DFEAT; ++k) {
    float h = hr[k];
    sr = fmaf(h, wr[k], sr);
    sv = fmaf(h, wv[k], sv);
  }
  out[i] = sr + br[0];
  out[ng + i] = sv + bv[0];
}

// -------------------------------------------------------------- gemm driver
static inline void launch_gemm(const float* Z, const float* W, const float* b,
                               float* Out, int rows, int relu, hipStream_t stream) {
  int rows16 = (rows / 16) * 16;
  if (rows16 > 0) {
    int gblocks = (rows16 / 16 + 7) / 8;
    gemm64_wmma<<<gblocks, 256, 0, stream>>>(Z, W, b, Out, rows16, relu);
  }
  int tail = rows - rows16;
  if (tail > 0) {
    gemm64_tail<<<(tail * DFEAT + 255) / 256, 256, 0, stream>>>(Z, W, b, Out,
                                                                rows16, rows, relu);
  }
}

// ---------------------------------------------------------------------------
extern "C" void kernel_launch(void* const* d_in, const int* in_sizes, int n_in,
                              void* d_out, int out_size, void* d_ws, size_t ws_size,
                              hipStream_t stream) {
  const float* x   = (const float*)d_in[0];
  const int* eidx  = (const int*)d_in[1];
  const int* n2s   = (const int*)d_in[2];
  const int* s2g   = (const int*)d_in[3];
  const float* cw1 = (const float*)d_in[4];
  const float* cb1 = (const float*)d_in[5];
  const float* cw2 = (const float*)d_in[6];
  const float* cb2 = (const float*)d_in[7];
  const float* fhw = (const float*)d_in[8];
  const float* fhb = (const float*)d_in[9];
  const float* frw = (const float*)d_in[10];
  const float* frb = (const float*)d_in[11];
  const float* fvw = (const float*)d_in[12];
  const float* fvb = (const float*)d_in[13];

  int nnodes = in_sizes[2];
  int nedges = in_sizes[1] / 2;
  int nsub   = in_sizes[3];
  int L      = in_sizes[5] / DFEAT;  // conv_b1 is [L,64]
  int ng     = out_size / 2;         // tuple (out, out_var), each [ng,1]
  int ngpad  = ((ng + 15) / 16) * 16;

  const int* esrc = eidx;
  const int* edst = eidx + nedges;

  // workspace layout (floats)
  float* bufA = (float*)d_ws;                       // h
  float* bufB = bufA + (size_t)nnodes * DFEAT;      // agg/z
  float* bufC = bufB + (size_t)nnodes * DFEAT;      // relu(z W1 + b1)
  float* sg   = bufC + (size_t)nnodes * DFEAT;      // subgraph pool
  float* g    = sg + (size_t)nsub * DFEAT;          // graph pool (padded rows)
  float* hid  = g + (size_t)ngpad * DFEAT;          // MLP hidden

  int nf4 = nnodes * (DFEAT / 4);
  copy_f4<<<(nf4 + 255) / 256, 256, 0, stream>>>((const float4*)x, (float4*)bufA, nf4);

  long long sthreads = (long long)nedges * 16;
  int sblocks = (int)((sthreads + 255) / 256);

  for (int l = 0; l < L; ++l) {
    copy_f4<<<(nf4 + 255) / 256, 256, 0, stream>>>((const float4*)bufA, (float4*)bufB, nf4);
    scatter_add<<<sblocks, 256, 0, stream>>>(bufA, esrc, edst, bufB, nedges);
    launch_gemm(bufB, cw1 + (size_t)l * DFEAT * DFEAT, cb1 + (size_t)l * DFEAT,
                bufC, nnodes, 1, stream);
    launch_gemm(bufC, cw2 + (size_t)l * DFEAT * DFEAT, cb2 + (size_t)l * DFEAT,
                bufA, nnodes, 0, stream);
  }

  int nsf = nsub * DFEAT;
  int ngf = ngpad * DFEAT;
  zero_f<<<(nsf + 255) / 256, 256, 0, stream>>>(sg, nsf);
  zero_f<<<(ngf + 255) / 256, 256, 0, stream>>>(g, ngf);
  seg_sum<<<(nnodes * DFEAT + 255) / 256, 256, 0, stream>>>(bufA, n2s, sg, nnodes);
  seg_sum<<<(nsf + 255) / 256, 256, 0, stream>>>(sg, s2g, g, nsub);

  launch_gemm(g, fhw, fhb, hid, ngpad, 1, stream);  // rows 500..511 are zero pad
  head<<<(ng + 255) / 256, 256, 0, stream>>>(hid, frw, frb, fvw, fvb, (float*)d_out, ng);
}